// BatchTopK_30391188586620
// MI455X (gfx1250) — compile-verified
//
#include <hip/hip_runtime.h>
#include <stdint.h>

// ---------------- problem constants ----------------
#define K_INIT      32
#define EMA_E       0.003f
#define TILE_ELEMS  4096               // 16 KB fp32 tile per TDM transfer
#define HIST_THREADS 256
#define NSUB        4                  // LDS histogram replication

// ---------------- workspace layout (u32 indices) ----------------
#define WS_HIST0     0                 // 2048 bins  (key bits 31:21)
#define WS_HIST1     2048              // 2048 bins  (key bits 20:10)
#define WS_HIST2     4096              // 1024 bins  (key bits  9:0)
#define WS_STATE     5120              // [0]=kRem [1]=keyPrefix [2]=thresholdKey [3]=tieQuota [4]=tieCounter
#define WS_TOTAL_U32 5136

typedef unsigned int v4u __attribute__((ext_vector_type(4)));
typedef int          v8i __attribute__((ext_vector_type(8)));
typedef int          v4i __attribute__((ext_vector_type(4)));
typedef float        f4  __attribute__((ext_vector_type(4)));
typedef float        v2f __attribute__((ext_vector_type(2)));
typedef float        v8f __attribute__((ext_vector_type(8)));

// order-preserving float -> uint key (larger float => larger key)
__device__ __forceinline__ unsigned key_of(float f) {
  unsigned u = __float_as_uint(f);
  return (u & 0x80000000u) ? ~u : (u | 0x80000000u);
}
__device__ __forceinline__ float val_of(unsigned k) {
  unsigned u = (k & 0x80000000u) ? (k & 0x7FFFFFFFu) : ~k;
  return __uint_as_float(u);
}

// ---------------- TDM: DMA one 1-row tile of nElem fp32 into LDS ----------------
__device__ __forceinline__ void tdm_load_tile(const float* gsrc, unsigned lds_byte_off, int nElem) {
  unsigned long long ga = (unsigned long long)(size_t)(const void*)gsrc;
  unsigned ne = (unsigned)nElem;
  // D# group 0: count=1 | lds_addr | global_addr[31:0] | global_addr[56:32] + type=2
  v4u g0;
  g0.x = 1u;
  g0.y = lds_byte_off;
  g0.z = (unsigned)(ga & 0xFFFFFFFFull);
  g0.w = (unsigned)((ga >> 32) & 0x1FFFFFFull) | (2u << 30);
  // D# group 1: data_size=4B, tensor_dim0=ne, tensor_dim1=1, tile_dim0=ne, tile_dim1=1
  v8i g1;
  g1[0] = (int)(2u << 16);                               // workgroup_mask=0, data_size=2 (4B)
  g1[1] = (int)((ne & 0xFFFFu) << 16);                   // barrier_addr=0 | tensor_dim0[15:0]
  g1[2] = (int)(((ne >> 16) & 0xFFFFu) | (1u << 16));    // tensor_dim0[31:16] | tensor_dim1[15:0]=1
  g1[3] = (int)((ne & 0xFFFFu) << 16);                   // tensor_dim1[31:16]=0 | tile_dim0
  g1[4] = 1;                                             // tile_dim1=1 | tile_dim2=0
  g1[5] = (int)ne;                                       // tensor_dim0_stride[31:0]
  g1[6] = (int)((ne & 0xFFFFu) << 16);                   // stride0[47:32]=0 | stride1[15:0]
  g1[7] = 0;                                             // stride1[47:16]=0
  v4i g2 = {0, 0, 0, 0};
  v4i g3 = {0, 0, 0, 0};
#if __clang_major__ >= 23
  v8i g4 = {0, 0, 0, 0, 0, 0, 0, 0};
  __builtin_amdgcn_tensor_load_to_lds(g0, g1, g2, g3, g4, 0);
#else
  __builtin_amdgcn_tensor_load_to_lds(g0, g1, g2, g3, 0);
#endif
}

// ---------------- init: zero histograms + state ----------------
__global__ void init_ws(unsigned* __restrict__ ws, int kTotal) {
  int i = blockIdx.x * blockDim.x + threadIdx.x;
  if (i < WS_TOTAL_U32) ws[i] = 0u;
  if (i == 0) ws[WS_STATE + 0] = (unsigned)kTotal;
}

// ---------------- pass 0: 2048-bin histogram of key[31:21], TDM-staged ----------------
__global__ void __launch_bounds__(HIST_THREADS)
hist_pass0(const float* __restrict__ src, unsigned* __restrict__ ws, int ntiles) {
  __shared__ unsigned lh[NSUB * 2048];
  __shared__ __align__(16) float stage[2][TILE_ELEMS];

  for (int i = threadIdx.x; i < NSUB * 2048; i += HIST_THREADS) lh[i] = 0u;

  unsigned lds0 = (unsigned)(size_t)(void*)&stage[0][0];   // flat addr low 32 bits = LDS offset
  unsigned lds1 = (unsigned)(size_t)(void*)&stage[1][0];
  unsigned sub  = (threadIdx.x & (NSUB - 1)) * 2048;

  int stride = gridDim.x;
  int t = blockIdx.x;
  if (threadIdx.x == 0 && t < ntiles)
    tdm_load_tile(src + (long long)t * TILE_ELEMS, lds0, TILE_ELEMS);
  __syncthreads();

  int buf = 0;
  for (; t < ntiles; t += stride, buf ^= 1) {
    if (threadIdx.x == 0) {
      int tn = t + stride;
      if (tn < ntiles) {
        tdm_load_tile(src + (long long)tn * TILE_ELEMS, buf ? lds0 : lds1, TILE_ELEMS);
        __builtin_amdgcn_s_wait_tensorcnt((short)1);   // oldest (current tile) complete
      } else {
        __builtin_amdgcn_s_wait_tensorcnt((short)0);
      }
    }
    __syncthreads();                                   // tile t visible to all waves
    const f4* s4 = reinterpret_cast<const f4*>(&stage[buf][0]);
#pragma unroll
    for (int i = threadIdx.x; i < TILE_ELEMS / 4; i += HIST_THREADS) {
      f4 v = s4[i];                                    // ds_load_b128
      atomicAdd(&lh[sub + (key_of(v.x) >> 21)], 1u);
      atomicAdd(&lh[sub + (key_of(v.y) >> 21)], 1u);
      atomicAdd(&lh[sub + (key_of(v.z) >> 21)], 1u);
      atomicAdd(&lh[sub + (key_of(v.w) >> 21)], 1u);
    }
    __syncthreads();                                   // all reads done before buf is re-DMA'd
  }

  unsigned* hist0 = ws + WS_HIST0;
  for (int b = threadIdx.x; b < 2048; b += HIST_THREADS) {
    unsigned c = 0;
#pragma unroll
    for (int s = 0; s < NSUB; ++s) c += lh[s * 2048 + b];
    if (c) atomicAdd(&hist0[b], c);
  }
}

// ---------------- refinement histogram (L2-resident pass) ----------------
__global__ void __launch_bounds__(256)
hist_refine(const float* __restrict__ src, unsigned* __restrict__ ws,
            unsigned maskHi, int shift, unsigned binMask, int histOff, int n4) {
  __shared__ unsigned lh[2048];
  for (int i = threadIdx.x; i < 2048; i += 256) lh[i] = 0u;
  __syncthreads();

  unsigned pref = ws[WS_STATE + 1] & maskHi;
  const f4* s4 = reinterpret_cast<const f4*>(src);
  int stride = gridDim.x * 256;
  for (int i = blockIdx.x * 256 + threadIdx.x; i < n4; i += stride) {
    __builtin_prefetch(&s4[i + stride], 0, 1);         // speculative global_prefetch
    f4 v = s4[i];
    unsigned k0 = key_of(v.x), k1 = key_of(v.y), k2 = key_of(v.z), k3 = key_of(v.w);
    if ((k0 & maskHi) == pref) atomicAdd(&lh[(k0 >> shift) & binMask], 1u);
    if ((k1 & maskHi) == pref) atomicAdd(&lh[(k1 >> shift) & binMask], 1u);
    if ((k2 & maskHi) == pref) atomicAdd(&lh[(k2 >> shift) & binMask], 1u);
    if ((k3 & maskHi) == pref) atomicAdd(&lh[(k3 >> shift) & binMask], 1u);
  }
  __syncthreads();

  unsigned* hist = ws + histOff;
  for (int b = threadIdx.x; b <= (int)binMask; b += 256) {
    unsigned c = lh[b];
    if (c) atomicAdd(&hist[b], c);
  }
}

// ---------------- pick cutoff bin from a histogram (single block) ----------------
// The 256-point suffix scan is done wave-wide as two chained triangular matmuls on
// V_WMMA_F32_16X16X4_F32: S1 = P x L (intra-row suffix), D = L x bcast(T) + S1 (inter-row).
__global__ void __launch_bounds__(256)
scan_pick(unsigned* __restrict__ ws, int histOff, int bins, int shift, int isFinal,
          const float* __restrict__ thrIn, float* __restrict__ thrOut) {
  __shared__ float partF[256];
  __shared__ float sufF[256];
  __shared__ float rowT[16];
  const unsigned* hist = ws + histOff;
  unsigned* st = ws + WS_STATE;
  int tid = threadIdx.x;
  unsigned kRem = st[0];                 // read before anyone can write (writes are post-barrier)
  int per = bins >> 8;
  int base = tid * per;
  unsigned s = 0;
  for (int j = 0; j < per; ++j) s += hist[base + j];
  partF[tid] = (float)s;
  __syncthreads();

  if (tid < 32) {                        // wave 0 only: EXEC all-ones for WMMA
    const int lane = tid;
    const int nrow = lane & 15;          // A: row M   |  B/C/D: column N
    const int grp  = lane >> 4;          // hi half-wave carries K+2 (A/B) and M+8 (C/D)
    const int koff = grp * 2;

    v8f acc = {0.f, 0.f, 0.f, 0.f, 0.f, 0.f, 0.f, 0.f};
    // Stage 1: S1[r][c] = sum_{c'>c} P[r][c'] ; P[r][c] = partF[16r+c]
#pragma unroll
    for (int k = 0; k < 4; ++k) {
      v2f a, b;
      a.x = partF[16 * nrow + 4 * k + koff];            // A[M][K]
      a.y = partF[16 * nrow + 4 * k + koff + 1];
      b.x = (4 * k + koff)     > nrow ? 1.0f : 0.0f;    // L[K][N] = [K > N]
      b.y = (4 * k + koff + 1) > nrow ? 1.0f : 0.0f;
      acc = __builtin_amdgcn_wmma_f32_16x16x4_f32(false, a, false, b, (short)0, acc,
                                                  false, false);
    }
#pragma unroll
    for (int v = 0; v < 8; ++v) sufF[16 * (v + 8 * grp) + nrow] = acc[v];  // [M][N]
    asm volatile("s_wait_dscnt 0x0" ::: "memory");
    if (lane < 16) rowT[lane] = sufF[16 * lane] + partF[16 * lane];        // full row sums
    asm volatile("s_wait_dscnt 0x0" ::: "memory");

    // Stage 2: D[r][c] = S1[r][c] + sum_{r'>r} T[r']
#pragma unroll
    for (int k = 0; k < 4; ++k) {
      v2f a, b;
      a.x = (4 * k + koff)     > nrow ? 1.0f : 0.0f;    // L[M][K] = [K > M]
      a.y = (4 * k + koff + 1) > nrow ? 1.0f : 0.0f;
      b.x = rowT[4 * k + koff];                         // T broadcast across columns
      b.y = rowT[4 * k + koff + 1];
      acc = __builtin_amdgcn_wmma_f32_16x16x4_f32(false, a, false, b, (short)0, acc,
                                                  false, false);
    }
#pragma unroll
    for (int v = 0; v < 8; ++v) sufF[16 * (v + 8 * grp) + nrow] = acc[v];
  }
  __syncthreads();

  // sums near the cutoff (<= ~2*kRem << 2^24) are exact integers in f32
  float aboveF = sufF[tid];
  float partS  = partF[tid];
  float kRemF  = (float)kRem;
  if (aboveF < kRemF && aboveF + partS >= kRemF) {     // cutoff lives in my chunk
    unsigned acc0 = (unsigned)(aboveF + 0.5f);
    int sel = base;
    unsigned a2 = acc0;
    for (int b = base + per - 1; b >= base; --b) {     // exact u32 intra-chunk scan
      unsigned c = hist[b];
      if (a2 + c >= kRem) { sel = b; break; }
      a2 += c;
    }
    unsigned rem = kRem - a2;
    unsigned kp = st[1] | ((unsigned)sel << shift);
    st[0] = rem;
    st[1] = kp;
    if (isFinal) {
      st[2] = kp;                        // exact threshold key (k-th largest)
      st[3] = rem;                       // tie quota at the threshold
      float v = val_of(kp);
      float mink = v > 0.0f ? v : 0.0f;  // min_k = relu(kth value)
      thrOut[0] = (1.0f - EMA_E) * thrIn[0] + EMA_E * mink;
    }
  }
}

// ---------------- final scatter: relu'd top-k, zeros elsewhere ----------------
__global__ void __launch_bounds__(256)
write_out(const float* __restrict__ src, float* __restrict__ dst, unsigned* ws, int n4) {
  unsigned keyT  = ws[WS_STATE + 2];
  unsigned quota = ws[WS_STATE + 3];
  unsigned* tie  = ws + WS_STATE + 4;
  const f4* s4 = reinterpret_cast<const f4*>(src);
  f4* d4 = reinterpret_cast<f4*>(dst);
  int stride = gridDim.x * 256;
  for (int i = blockIdx.x * 256 + threadIdx.x; i < n4; i += stride) {
    f4 v = s4[i];
    f4 o;
    {
      unsigned k = key_of(v.x);
      o.x = (k > keyT || (k == keyT && atomicAdd(tie, 1u) < quota)) ? fmaxf(v.x, 0.0f) : 0.0f;
    }
    {
      unsigned k = key_of(v.y);
      o.y = (k > keyT || (k == keyT && atomicAdd(tie, 1u) < quota)) ? fmaxf(v.y, 0.0f) : 0.0f;
    }
    {
      unsigned k = key_of(v.z);
      o.z = (k > keyT || (k == keyT && atomicAdd(tie, 1u) < quota)) ? fmaxf(v.z, 0.0f) : 0.0f;
    }
    {
      unsigned k = key_of(v.w);
      o.w = (k > keyT || (k == keyT && atomicAdd(tie, 1u) < quota)) ? fmaxf(v.w, 0.0f) : 0.0f;
    }
    __builtin_nontemporal_store(o, &d4[i]);            // write-once output: keep input in L2
  }
}

extern "C" void kernel_launch(void* const* d_in, const int* in_sizes, int n_in,
                              void* d_out, int out_size, void* d_ws, size_t ws_size,
                              hipStream_t stream) {
  const float* feat = (const float*)d_in[0];
  const float* thr  = (const float*)d_in[1];
  float* out = (float*)d_out;
  unsigned* ws = (unsigned*)d_ws;

  const int N      = in_sizes[0];        // 512*6*8192 = 25,165,824
  const int n4     = N / 4;
  const int ntiles = N / TILE_ELEMS;     // 6144
  const int kTotal = K_INIT * (N / 8192);// 32 * 512 * 6 = 98,304
  float* thrOut = out + N;               // d_out = [out(N) | new_threshold(1)]

  init_ws<<<(WS_TOTAL_U32 + 255) / 256, 256, 0, stream>>>(ws, kTotal);

  // pass 0: HBM -> LDS via TDM, histogram of key[31:21]
  hist_pass0<<<1024, HIST_THREADS, 0, stream>>>(feat, ws, ntiles);
  scan_pick<<<1, 256, 0, stream>>>(ws, WS_HIST0, 2048, 21, 0, thr, thrOut);

  // pass 1: L2-resident refine, key[20:10]
  hist_refine<<<2048, 256, 0, stream>>>(feat, ws, 0xFFE00000u, 10, 2047u, WS_HIST1, n4);
  scan_pick<<<1, 256, 0, stream>>>(ws, WS_HIST1, 2048, 10, 0, thr, thrOut);

  // pass 2: L2-resident refine, key[9:0] -> exact threshold + EMA update
  hist_refine<<<2048, 256, 0, stream>>>(feat, ws, 0xFFFFFC00u, 0, 1023u, WS_HIST2, n4);
  scan_pick<<<1, 256, 0, stream>>>(ws, WS_HIST2, 1024, 0, 1, thr, thrOut);

  // final: scatter relu'd top-k with exact-k tie handling
  write_out<<<2048, 256, 0, stream>>>(feat, out, ws, n4);
}